// MultiHeadDistanceKNN_20633022890611
// MI455X (gfx1250) — compile-verified
//
#include <hip/hip_runtime.h>
#include <hip/hip_bf16.h>

typedef __attribute__((ext_vector_type(2))) float v2f;
typedef __attribute__((ext_vector_type(8))) float v8f;

#define BDIM  2
#define NDIM  2048
#define DDIM  512
#define HEADS 4
#define HD    128
#define KSEL  307            // int(2048 * 0.15)
#define SSP   (NDIM + 4)     // k_stats LDS row stride (bank-conflict pad)
#define JCH   1024           // k_adj columns per block
#define ASP   (JCH + 4)      // k_adj LDS row stride (bank-conflict pad)

__device__ __forceinline__ v8f wmma_f32_k4(v2f a, v2f b, v8f c) {
  // D = A(16x4,f32) * B(4x16,f32) + C(16x16,f32), exact fp32 matrix pipe
  return __builtin_amdgcn_wmma_f32_16x16x4_f32(false, a, false, b, (short)0, c,
                                               false, false);
}

// ---------------------------------------------------------------------------
// z[h, b*N+n, :] = x[b*N+n, :] @ W[h].  One wave computes a full 16x128 row
// tile (8 accumulators) so each A-fragment load feeds 8 WMMAs.
// 1024 wave-tasks -> 128 blocks x 8 waves.
// ---------------------------------------------------------------------------
__global__ __launch_bounds__(256) void k_project(const float* __restrict__ x,
                                                 const float* __restrict__ W,
                                                 float* __restrict__ z) {
  const int lane  = threadIdx.x & 31;
  const int wave  = threadIdx.x >> 5;
  const int t     = blockIdx.x * 8 + wave;      // 256 row tiles per head
  const int h     = t >> 8;
  const int m0    = (t & 255) << 4;             // row tile in [0, 4096)
  const int l15   = lane & 15;
  const int khalf = (lane >> 4) << 1;           // 0 for lanes 0-15, 2 for 16-31
  const int mhi   = (lane >> 4) << 3;

  const float* Wh   = W + (size_t)h * DDIM * HD;
  const float* arow = x + (size_t)(m0 + l15) * DDIM + khalf;

  v8f acc[8];
#pragma unroll
  for (int nt = 0; nt < 8; ++nt) acc[nt] = (v8f){0.f,0.f,0.f,0.f,0.f,0.f,0.f,0.f};

  for (int k = 0; k < DDIM; k += 4) {
    const v2f a = *(const v2f*)(arow + k);
    const int kk = k + khalf;
#pragma unroll
    for (int nt = 0; nt < 8; ++nt) {
      v2f bb;
      bb.x = Wh[(size_t)kk * HD + (nt << 4) + l15];
      bb.y = Wh[(size_t)(kk + 1) * HD + (nt << 4) + l15];
      acc[nt] = wmma_f32_k4(a, bb, acc[nt]);
    }
  }

  float* zt = z + (size_t)h * BDIM * NDIM * HD;
#pragma unroll
  for (int nt = 0; nt < 8; ++nt)
#pragma unroll
    for (int r = 0; r < 8; ++r)
      zt[(size_t)(m0 + mhi + r) * HD + (nt << 4) + l15] = acc[nt][r];
}

// ---------------------------------------------------------------------------
// sq[row] = ||z_row||^2   for all H*B*N rows
// ---------------------------------------------------------------------------
__global__ __launch_bounds__(256) void k_sqnorm(const float* __restrict__ z,
                                                float* __restrict__ sq) {
  const int row = blockIdx.x * blockDim.x + threadIdx.x;
  const float4* zr = (const float4*)(z + (size_t)row * HD);
  float s = 0.f;
#pragma unroll
  for (int i = 0; i < HD / 4; ++i) {
    float4 v = zr[i];
    s += v.x * v.x + v.y * v.y + v.z * v.z + v.w * v.w;
  }
  sq[row] = s;
}

__global__ void k_init(float* __restrict__ hsum) {
  if (threadIdx.x < HEADS) hsum[threadIdx.x] = 0.f;
}

// ---------------------------------------------------------------------------
// For each (h, b, 16-row strip): WMMA the 16x2048 distance strip into LDS
// (~128 KB of the WGP's 320 KB), accumulate sum(dist) -> hsum[h], then find
// each row's KSEL-th smallest distance via bitwise binary search on the float
// bit pattern (non-negative floats order as uints). 1024 blocks.
// A-fragments (64 VGPRs) are cached across all 16 column tiles per wave.
// ---------------------------------------------------------------------------
__global__ __launch_bounds__(256) void k_stats(const float* __restrict__ z,
                                               const float* __restrict__ sq,
                                               float* __restrict__ thresh,
                                               float* __restrict__ hsum) {
  extern __shared__ float smem[];               // 16*SSP dist strip + 8 partials
  float* dist_s = smem;
  float* red    = smem + 16 * SSP;

  const int lane  = threadIdx.x & 31;
  const int wave  = threadIdx.x >> 5;
  const int bid   = blockIdx.x;                 // H * B * (N/16)
  const int h     = bid >> 8;                   // 256 blocks per head
  const int rem   = bid & 255;
  const int b     = rem >> 7;
  const int i0    = (rem & 127) << 4;
  const int l15   = lane & 15;
  const int khalf = (lane >> 4) << 1;
  const int mhi   = (lane >> 4) << 3;

  const float* zp  = z  + (size_t)(h * BDIM + b) * NDIM * HD;
  const float* sqp = sq + (size_t)(h * BDIM + b) * NDIM;

  // cache A-fragments for the 16-row strip (reused by all column tiles)
  const float* ar = zp + (size_t)(i0 + l15) * HD + khalf;
  v2f afr[HD / 4];
#pragma unroll
  for (int k = 0; k < HD; k += 4) afr[k >> 2] = *(const v2f*)(ar + k);

  float sqi[8];
#pragma unroll
  for (int r = 0; r < 8; ++r) sqi[r] = sqp[i0 + mhi + r];

  float lsum = 0.f;
  for (int jt = wave; jt < NDIM / 16; jt += 8) {
    const int j0 = jt << 4;
    const float* br = zp + (size_t)(j0 + l15) * HD + khalf;
    __builtin_prefetch(br + (size_t)128 * HD, 0, 3);   // next tile for this wave
    v8f acc = {0.f, 0.f, 0.f, 0.f, 0.f, 0.f, 0.f, 0.f};
#pragma unroll
    for (int k = 0; k < HD; k += 4)
      acc = wmma_f32_k4(afr[k >> 2], *(const v2f*)(br + k), acc);

    const float sqj = sqp[j0 + l15];
#pragma unroll
    for (int r = 0; r < 8; ++r) {
      float d2 = __builtin_fmaf(-2.f, acc[r], sqi[r] + sqj);
      float d  = sqrtf(fmaxf(d2, 0.f));
      dist_s[(mhi + r) * SSP + j0 + l15] = d;
      lsum += d;
    }
  }

  // block-reduce sum(dist) -> hsum[h]
  for (int off = 16; off; off >>= 1) lsum += __shfl_xor(lsum, off, 32);
  if (lane == 0) red[wave] = lsum;
  __syncthreads();                               // strip fully written, too
  if (wave == 0) {
    float s = (lane < 8) ? red[lane] : 0.f;
    for (int off = 16; off; off >>= 1) s += __shfl_xor(s, off, 32);
    if (lane == 0) atomicAdd(&hsum[h], s);
  }

  // per-row exact k-th smallest (one wave per row, 2 rows per wave)
  for (int row = wave; row < 16; row += 8) {
    const float* dr = dist_s + row * SSP;
    unsigned lo = 0u, hi = 0x7f800000u;
    while (lo < hi) {
      const unsigned mid = (lo + hi) >> 1;
      int c = 0;
      for (int col = lane; col < NDIM; col += 32)
        c += (__float_as_uint(dr[col]) <= mid) ? 1 : 0;
      for (int off = 16; off; off >>= 1) c += __shfl_xor(c, off, 32);
      if (c >= KSEL) hi = mid; else lo = mid + 1;
    }
    if (lane == 0)
      thresh[(size_t)(h * BDIM + b) * NDIM + i0 + row] = __uint_as_float(hi);
  }
}

// ---------------------------------------------------------------------------
// adj[b,i,j] = (1/H) * sum_h exp(-d^2/(2*mean_h^2+1e-8)) * [d <= max(ti,tj)]
// One block per (b, 16-row strip, 1024-col chunk); heads loop OUTERMOST so the
// cached A-fragments are reused over 64 column tiles per head; the head sum is
// accumulated in a 64 KB LDS strip and written out coalesced. The K-loop is
// the identical WMMA sequence as k_stats -> bit-identical distances.
// ---------------------------------------------------------------------------
__global__ __launch_bounds__(256) void k_adj(const float* __restrict__ z,
                                             const float* __restrict__ sq,
                                             const float* __restrict__ thresh,
                                             const float* __restrict__ hsum,
                                             float* __restrict__ out) {
  extern __shared__ float accs[];               // 16 * ASP
  const int lane  = threadIdx.x & 31;
  const int wave  = threadIdx.x >> 5;
  const int tid   = threadIdx.x;
  const int bid   = blockIdx.x;                 // B * 128 * 2 = 512
  const int b     = bid >> 8;
  const int rem   = bid & 255;
  const int i0    = (rem >> 1) << 4;
  const int jbase = (rem & 1) * JCH;
  const int l15   = lane & 15;
  const int khalf = (lane >> 4) << 1;
  const int mhi   = (lane >> 4) << 3;
  const float inv_cnt = 1.0f / (float)((size_t)BDIM * NDIM * NDIM);

  for (int idx = tid; idx < 16 * ASP; idx += 256) accs[idx] = 0.f;
  __syncthreads();

  for (int h = 0; h < HEADS; ++h) {
    const float* zp  = z      + (size_t)(h * BDIM + b) * NDIM * HD;
    const float* sqp = sq     + (size_t)(h * BDIM + b) * NDIM;
    const float* thp = thresh + (size_t)(h * BDIM + b) * NDIM;
    const float mean  = hsum[h] * inv_cnt;
    const float inv2s = 1.0f / (2.f * mean * mean + 1e-8f);

    const float* ar = zp + (size_t)(i0 + l15) * HD + khalf;
    v2f afr[HD / 4];
#pragma unroll
    for (int k = 0; k < HD; k += 4) afr[k >> 2] = *(const v2f*)(ar + k);

    float sqi[8], ti[8];
#pragma unroll
    for (int r = 0; r < 8; ++r) {
      sqi[r] = sqp[i0 + mhi + r];
      ti[r]  = thp[i0 + mhi + r];
    }

    for (int jt = wave; jt < JCH / 16; jt += 8) {
      const int j0 = jbase + (jt << 4);
      const float* br = zp + (size_t)(j0 + l15) * HD + khalf;
      __builtin_prefetch(br + (size_t)128 * HD, 0, 3);
      v8f acc = {0.f, 0.f, 0.f, 0.f, 0.f, 0.f, 0.f, 0.f};
#pragma unroll
      for (int k = 0; k < HD; k += 4)
        acc = wmma_f32_k4(afr[k >> 2], *(const v2f*)(br + k), acc);

      const float sqj = sqp[j0 + l15];
      const float tj  = thp[j0 + l15];
      const int   cb  = j0 - jbase + l15;
#pragma unroll
      for (int r = 0; r < 8; ++r) {
        float d2 = __builtin_fmaf(-2.f, acc[r], sqi[r] + sqj);
        float d  = sqrtf(fmaxf(d2, 0.f));
        if (d <= fmaxf(ti[r], tj))
          accs[(mhi + r) * ASP + cb] += expf(-d * d * inv2s);
      }
    }
  }
  __syncthreads();

  float* ob = out + (size_t)b * NDIM * NDIM + (size_t)i0 * NDIM + jbase;
  for (int idx = tid; idx < 16 * JCH; idx += 256) {
    const int row = idx >> 10;                   // JCH == 1024
    const int col = idx & (JCH - 1);
    ob[(size_t)row * NDIM + col] = 0.25f * accs[row * ASP + col];
  }
}

extern "C" void kernel_launch(void* const* d_in, const int* in_sizes, int n_in,
                              void* d_out, int out_size, void* d_ws, size_t ws_size,
                              hipStream_t stream) {
  (void)in_sizes; (void)n_in; (void)out_size; (void)ws_size;
  const float* x = (const float*)d_in[0];        // [B, N, D]
  const float* W = (const float*)d_in[1];        // [H, D, hd]
  float* out = (float*)d_out;                    // [B, N, N]

  // workspace layout (floats): z | sq | thresh | hsum   (~8.5 MB, L2-resident)
  float* z      = (float*)d_ws;                              // H*B*N*HD
  float* sq     = z + (size_t)HEADS * BDIM * NDIM * HD;      // H*B*N
  float* thresh = sq + (size_t)HEADS * BDIM * NDIM;          // H*B*N
  float* hsum   = thresh + (size_t)HEADS * BDIM * NDIM;      // H

  k_init<<<1, 64, 0, stream>>>(hsum);
  k_project<<<128, 256, 0, stream>>>(x, W, z);
  k_sqnorm<<<(HEADS * BDIM * NDIM) / 256, 256, 0, stream>>>(z, sq);
  const size_t smem_stats = (size_t)(16 * SSP + 8) * sizeof(float);  // ~128 KB
  k_stats<<<HEADS * BDIM * (NDIM / 16), 256, smem_stats, stream>>>(z, sq, thresh,
                                                                   hsum);
  const size_t smem_adj = (size_t)(16 * ASP) * sizeof(float);        // ~64 KB
  k_adj<<<BDIM * (NDIM / 16) * (NDIM / JCH), 256, smem_adj, stream>>>(z, sq,
                                                                      thresh,
                                                                      hsum, out);
}